// SequenceGroupAggregating_28965259444567
// MI455X (gfx1250) — compile-verified
//
#include <hip/hip_runtime.h>

// SequenceGroupAggregating (segment mean over sorted indexes) for MI455X/gfx1250.
//
// out[b,t,:] = sum_{s: idx[b,s]==t} hidden[b,s,:] / max(count_t, 1)
//
// Bandwidth-bound: ~75.6 MB total traffic -> >= 3.2us at 23.3 TB/s.
// Full f32 (downconversion saves nothing on a BW-bound f32 stream). The
// segment-sum is a sorted-sparsity-restricted mask bmm on
// V_WMMA_F32_16X16X4_F32: A = 16x4 one-hot mask built in-register from index
// compares, B = 4x16 tile of hidden, C/D = 16 tokens x 16 features f32 acc.
// Sorted indexes => each 16-token tile only needs a contiguous K-range of
// rows (binary search), so WMMA count is tiny vs the memory floor.
//
// v2: branch-free A-mask -> one load clause + single staged loadcnt wait per
//     K-step, no divergent EXEC juggling.
// v3: epilogue reciprocals via v_rcp_f32 (__builtin_amdgcn_rcpf) instead of
//     the IEEE div_scale/fma expansion — ~60 fewer VALU ops per block in a
//     kernel whose K-loop is only ~10 iterations.

typedef __attribute__((ext_vector_type(2))) float v2f;
typedef __attribute__((ext_vector_type(8))) float v8f;

#define NB   8      // batch
#define NS   2048   // sub-tokens per batch
#define ND   768    // hidden dim
#define NT   1024   // output tokens per batch
#define NCHUNK 128  // features handled per block (4 waves x 2 n-tiles x 16)

__global__ __launch_bounds__(128)
void seq_group_agg_wmma(const float* __restrict__ hidden,
                        const int*   __restrict__ ori_idx,  // int32 per harness
                        float*       __restrict__ out)
{
    __shared__ int sPos[17];   // row positions of tokens t0..t0+16 (lower bounds)

    const int nPerTile = ND / NCHUNK;                 // 6 feature chunks
    const int bid  = blockIdx.x;
    const int b    = bid / (64 * nPerTile);
    const int rem  = bid % (64 * nPerTile);
    const int tile = rem / nPerTile;                  // 16-token tile id (0..63)
    const int nblk = rem % nPerTile;                  // feature chunk (0..5)
    const int t0   = tile * 16;
    const int nbase = nblk * NCHUNK;

    const int* idx_b = ori_idx + b * NS;
    const int  tid   = threadIdx.x;

    // --- lower_bound for tokens t0 .. t0+16 over the sorted index array ---
    if (tid < 17) {
        const int target = t0 + tid;
        int lo = 0, hi = NS;
        while (lo < hi) {
            int mid = (lo + hi) >> 1;
            if (idx_b[mid] < target) lo = mid + 1; else hi = mid;
        }
        sPos[tid] = lo;
    }
    __syncthreads();

    const int lo    = sPos[0];
    const int hiEnd = sPos[16];

    const int lane = tid & 31;
    const int wave = tid >> 5;        // 0..3
    const int lm   = lane & 15;       // N (and A's M) index within tile
    const int hh   = lane >> 4;       // lane half: selects K/M offset
    const int tlm  = t0 + lm;

    // Each wave owns two adjacent 16-wide feature tiles.
    const int n0 = nbase + wave * 32;

    v8f acc0 = {};   // features [n0,    n0+16)
    v8f acc1 = {};   // features [n0+16, n0+32)

    const size_t rowBase = (size_t)b * NS * ND;

    for (int kb = (lo & ~3); kb < hiEnd; kb += 4) {
        const int k0 = kb + 2 * hh;
        const int k1 = k0 + 1;
        const int r0 = min(k0, NS - 1);
        const int r1 = min(k1, NS - 1);

        // ---- all 6 loads of the iteration, unconditional & in-bounds ----
        const int vi0 = idx_b[r0];
        const int vi1 = idx_b[r1];
        const float* p0 = hidden + rowBase + (size_t)r0 * ND;  // rows K=0 / K=2
        const float* p1 = hidden + rowBase + (size_t)r1 * ND;  // rows K=1 / K=3
        v2f b0, b1;
        b0.x = p0[n0 + lm];       b0.y = p1[n0 + lm];
        b1.x = p0[n0 + 16 + lm];  b1.y = p1[n0 + 16 + lm];

        // Prefetch next K-chunk of this stream (emits global_prefetch_b8).
        __builtin_prefetch(hidden + rowBase + (size_t)min(r0 + 4, NS - 1) * ND + n0 + lm, 0, 1);

        // ---- A fragment: one-hot mask, 16x4 f32 in 2 VGPRs (branch-free) ----
        // lane (lm, hh): a.x = (M=lm, K=2*hh), a.y = (M=lm, K=2*hh+1)
        // Rows in [kb&~3, lo) have idx < t0 (sorted) so the compare already
        // zeroes them; rows >= hiEnd are zeroed by the explicit predicate.
        v2f a;
        a.x = (((int)(k0 < hiEnd) & (int)(vi0 == tlm)) != 0) ? 1.0f : 0.0f;
        a.y = (((int)(k1 < hiEnd) & (int)(vi1 == tlm)) != 0) ? 1.0f : 0.0f;

        acc0 = __builtin_amdgcn_wmma_f32_16x16x4_f32(
                   false, a, false, b0, (short)0, acc0, false, false);
        acc1 = __builtin_amdgcn_wmma_f32_16x16x4_f32(
                   false, a, false, b1, (short)0, acc1, false, false);
    }

    // ---- mean: per-token reciprocal counts (clamped >= 1) ----
    // v_rcp_f32 (~1 ulp) instead of full IEEE division: counts are small
    // integers, the mean of ~2 values tolerates 1-ulp reciprocal error.
    float inv[8];
#pragma unroll
    for (int v = 0; v < 8; ++v) {
        const int t = 8 * hh + v;                   // token row this lane holds in VGPR v
        const int c = sPos[t + 1] - sPos[t];
        inv[v] = __builtin_amdgcn_rcpf((float)max(c, 1));
    }

    // ---- store: C layout -> lane lm gives N, VGPR v gives M = v + 8*hh ----
    float* outB = out + ((size_t)b * NT + t0) * ND;
#pragma unroll
    for (int v = 0; v < 8; ++v) {
        const int t = v + 8 * hh;
        outB[(size_t)t * ND + n0 + lm]      = acc0[v] * inv[v];
        outB[(size_t)t * ND + n0 + 16 + lm] = acc1[v] * inv[v];
    }
}

extern "C" void kernel_launch(void* const* d_in, const int* in_sizes, int n_in,
                              void* d_out, int out_size, void* d_ws, size_t ws_size,
                              hipStream_t stream) {
    const float* hidden = (const float*)d_in[0];
    const int*   idx    = (const int*)d_in[1];   // integer input -> const int*
    float*       out    = (float*)d_out;

    (void)in_sizes; (void)n_in; (void)out_size; (void)d_ws; (void)ws_size;

    // 8 batches x 64 token-tiles x 6 feature-chunks = 3072 blocks, 4 waves each.
    const int nBlocks = NB * (NT / 16) * (ND / NCHUNK);
    seq_group_agg_wmma<<<nBlocks, 128, 0, stream>>>(hidden, idx, out);
}